// SparseKANLayer_53807350284397
// MI455X (gfx1250) — compile-verified
//
#include <hip/hip_runtime.h>
#include <hip/hip_bf16.h>
#include <stdint.h>

#define IN_F      8192
#define OUT_F     8192
#define NGRID     8
#define BATCH     16
#define WCHUNK    64          // edges staged per wave per iteration
#define WAVES_PB  8
#define THREADS   256
#define BLOCKS    512

typedef unsigned int v4u __attribute__((ext_vector_type(4)));
typedef int          v8i __attribute__((ext_vector_type(8)));
typedef int          v4i __attribute__((ext_vector_type(4)));

// ---------------------------------------------------------------------------
// TDM: 1-D contiguous copy of ndw dwords from global -> LDS (async, TENSORcnt)
// D# per CDNA5 ISA ch.8: group0 = {count/flags, lds_addr, global_addr, type=2},
// group1 = {data_size=4B, tensor_dim0=ndw, tensor_dim1=1, tile_dim0=ndw, 1-D}.
// ---------------------------------------------------------------------------
__device__ __forceinline__ void tdm_copy_to_lds(uint32_t lds_off,
                                                const void* gsrc,
                                                uint32_t ndw) {
  uint64_t ga = (uint64_t)(uintptr_t)gsrc;
  v4u g0;
  g0[0] = 1u;                                            // count=1, user mode
  g0[1] = lds_off;                                       // lds_addr (bytes)
  g0[2] = (uint32_t)ga;                                  // global_addr[31:0]
  g0[3] = (uint32_t)((ga >> 32) & 0x01FFFFFFu) | (2u << 30); // addr[56:32] | type=2
  v8i g1;
  g1[0] = 0x00020000;                                    // data_size=2 (4 bytes)
  g1[1] = (int)((ndw & 0xFFFFu) << 16);                  // tensor_dim0[15:0]
  g1[2] = (int)(((ndw >> 16) & 0xFFFFu) | (1u << 16));   // tdim0[31:16] | tensor_dim1=1
  g1[3] = (int)((ndw & 0xFFFFu) << 16);                  // tile_dim0 = ndw
  g1[4] = 0;                                             // tile_dim1/2 unused -> 1-D
  g1[5] = (int)ndw;                                      // tensor_dim0_stride (unused, 1-D)
  g1[6] = 0;
  g1[7] = 0;
  v4i z4; z4[0] = z4[1] = z4[2] = z4[3] = 0;
#if __clang_major__ >= 23
  v8i z8; z8[0]=z8[1]=z8[2]=z8[3]=z8[4]=z8[5]=z8[6]=z8[7]=0;
  __builtin_amdgcn_tensor_load_to_lds(g0, g1, z4, z4, z8, 0);
#else
  __builtin_amdgcn_tensor_load_to_lds(g0, g1, z4, z4, 0);
#endif
}

__global__ __launch_bounds__(THREADS)
void kan_main_53807350284397(const float* __restrict__ x,
                             const int*   __restrict__ rows,
                             const int*   __restrict__ cols,
                             const float* __restrict__ wsp,
                             const float* __restrict__ wb,
                             const float* __restrict__ grid,
                             float* __restrict__ out,
                             int nE) {
  __shared__ __attribute__((aligned(16)))
      float lds_ws[WAVES_PB * 2 * WCHUNK * NGRID];        // 32 KB / block

  const int tid  = threadIdx.x;
  const int lane = tid & 31;
  const int wv   = tid >> 5;
  const int gwave      = (int)blockIdx.x * WAVES_PB + wv;
  const int totalWaves = (int)gridDim.x * WAVES_PB;
  const int nChunks    = (nE + WCHUNK - 1) / WCHUNK;

  // exp(-t^2) = exp2(-(t*sqrt(log2 e))^2), t = (xv - g) * (1/DENOM), DENOM = 4/7
  const float L2E = 1.4426950408889634f;
  const float C   = 1.75f * 1.2011224087864498f;          // invD * sqrt(L2E)
  float gc[NGRID];
#pragma unroll
  for (int g = 0; g < NGRID; ++g) gc[g] = grid[g] * C;    // uniform scalar loads

  const uint32_t waveLds  = (uint32_t)(uintptr_t)&lds_ws[wv * 2 * WCHUNK * NGRID];
  const uint32_t bufBytes = WCHUNK * NGRID * 4;

  int c = gwave;
  if (c < nChunks) {                                      // prologue: stage chunk 0
    uint32_t n = (uint32_t)min(WCHUNK, nE - c * WCHUNK) * NGRID;
    tdm_copy_to_lds(waveLds, wsp + (size_t)c * WCHUNK * NGRID, n);
  }

  int it = 0;
  for (; c < nChunks; c += totalWaves, ++it) {
    const int cn = c + totalWaves;
    asm volatile("" ::: "memory");
    if (cn < nChunks) {                                   // stage next into other buf
      uint32_t n = (uint32_t)min(WCHUNK, nE - cn * WCHUNK) * NGRID;
      tdm_copy_to_lds(waveLds + ((it + 1) & 1) * bufBytes,
                      wsp + (size_t)cn * WCHUNK * NGRID, n);
      __builtin_amdgcn_s_wait_tensorcnt(1);               // current chunk landed
    } else {
      __builtin_amdgcn_s_wait_tensorcnt(0);
    }
    asm volatile("" ::: "memory");                        // TDM wrote LDS: no folding

    const float4* pbuf =
        (const float4*)&lds_ws[(wv * 2 + (it & 1)) * WCHUNK * NGRID];

#pragma unroll
    for (int k = 0; k < WCHUNK / 32; ++k) {
      const int el = lane + k * 32;
      const int e  = c * WCHUNK + el;
      if (e < nE) {
        const int   row = __builtin_nontemporal_load(rows + e);
        const int   col = __builtin_nontemporal_load(cols + e);
        const float bw  = __builtin_nontemporal_load(wb + e);
        const float4 wlo = pbuf[el * 2];
        const float4 whi = pbuf[el * 2 + 1];
        const float ws[NGRID] = {wlo.x, wlo.y, wlo.z, wlo.w,
                                 whi.x, whi.y, whi.z, whi.w};
        const float* xcol = x + col;                      // L2-resident gathers
        float* orow = out + row;

        float xv[BATCH];
#pragma unroll
        for (int b = 0; b < BATCH; ++b)                   // 16 gathers in flight
          xv[b] = xcol[b << 13];

#pragma unroll
        for (int b = 0; b < BATCH; ++b) {
          const float v = xv[b];
          float acc = 0.0f;
#pragma unroll
          for (int g = 0; g < NGRID; ++g) {
            const float t = __builtin_fmaf(v, C, -gc[g]);
            const float r = __builtin_amdgcn_exp2f(-t * t);
            acc = __builtin_fmaf(r, ws[g], acc);
          }
          // silu(v) = v / (1 + exp(-v))
          const float ex  = __builtin_amdgcn_exp2f(-v * L2E);
          const float sig = __builtin_amdgcn_rcpf(1.0f + ex);
          const float contrib = __builtin_fmaf(v * sig, bw, acc);
          unsafeAtomicAdd(orow + (b << 13), contrib);     // global_atomic_add_f32
        }
      }
    }
  }
}

__global__ void kan_zero_53807350284397(float* __restrict__ out, int n) {
  int i = blockIdx.x * blockDim.x + threadIdx.x;
  if (i < n) out[i] = 0.0f;
}

extern "C" void kernel_launch(void* const* d_in, const int* in_sizes, int n_in,
                              void* d_out, int out_size, void* d_ws, size_t ws_size,
                              hipStream_t stream) {
  const float* x    = (const float*)d_in[0];
  const int*   rows = (const int*)d_in[1];
  const int*   cols = (const int*)d_in[2];
  const float* wsp  = (const float*)d_in[3];
  const float* wb   = (const float*)d_in[4];
  const float* grid = (const float*)d_in[5];
  float* out = (float*)d_out;
  const int nE = in_sizes[1];

  kan_zero_53807350284397<<<(out_size + 255) / 256, 256, 0, stream>>>(out, out_size);
  kan_main_53807350284397<<<BLOCKS, THREADS, 0, stream>>>(x, rows, cols, wsp, wb,
                                                          grid, out, nE);
}